// OCRead_12584254177707
// MI455X (gfx1250) — compile-verified
//
#include <hip/hip_runtime.h>
#include <stdint.h>

// Problem constants (match the JAX reference)
constexpr int Bb = 32;     // batch
constexpr int R  = 8192;   // rows per batch
constexpr int E  = 256;    // embedding dim
constexpr int K  = 64;     // clusters

constexpr int CHUNKS         = 16;            // row-chunks per batch (split-K for GEMM2)
constexpr int ROWS_PER_CHUNK = R / CHUNKS;    // 512
constexpr int TILE_ROWS      = 64;            // rows staged per LDS tile
constexpr int TILE_ITERS     = ROWS_PER_CHUNK / TILE_ROWS; // 8
constexpr int PSTR           = 65;            // padded row stride for p in LDS (floats)

typedef __attribute__((ext_vector_type(2))) float v2f;
typedef __attribute__((ext_vector_type(8))) float v8f;

// The async builtin's parameter type (from the round-2 diagnostic):
// pointer to 16-byte int vector, global AS for src / LDS AS for dst.
typedef int v4i __attribute__((vector_size(16)));
typedef __attribute__((address_space(1))) v4i gv4i_t;  // global
typedef __attribute__((address_space(3))) v4i lv4i_t;  // LDS

// D = A(16x4 f32) * B(4x16 f32) + C(16x16 f32), full-precision CDNA5 WMMA.
__device__ __forceinline__ v8f wmma_f32(v2f a, v2f b, v8f c) {
  return __builtin_amdgcn_wmma_f32_16x16x4_f32(
      /*neg_a=*/false, a, /*neg_b=*/false, b,
      /*c_mod=*/(short)0, c, /*reuse_a=*/false, /*reuse_b=*/false);
}

// 16B global -> LDS async copy (ASYNCcnt-tracked, no VGPR round trip).
__device__ __forceinline__ void async_b128(const float* g, float* l) {
#if __has_builtin(__builtin_amdgcn_global_load_async_to_lds_b128)
  __builtin_amdgcn_global_load_async_to_lds_b128((gv4i_t*)g, (lv4i_t*)l, 0, 0);
#else
  unsigned lds_addr = (unsigned)(uintptr_t)l;          // low 32 bits = LDS offset
  unsigned long long gaddr = (unsigned long long)(uintptr_t)g;
  asm volatile("global_load_async_to_lds_b128 %0, %1, off"
               :: "v"(lds_addr), "v"(gaddr) : "memory");
#endif
}

__device__ __forceinline__ void wait_async0() {
#if __has_builtin(__builtin_amdgcn_s_wait_asynccnt)
  __builtin_amdgcn_s_wait_asynccnt(0);
#else
  asm volatile("s_wait_asynccnt 0x0" ::: "memory");
#endif
}

// LDS layout (floats):
//   sC  : centers   [K=64][E=256]          (16384)
//   sX0 : x tile A  [TILE_ROWS=64][E=256]  (16384)
//   sX1 : x tile B  [TILE_ROWS=64][E=256]  (16384)
//   sP  : scores/p  [TILE_ROWS=64][PSTR]   (4160)
constexpr int LDS_FLOATS = K * E + 2 * TILE_ROWS * E + TILE_ROWS * PSTR;
constexpr size_t LDS_BYTES = (size_t)LDS_FLOATS * sizeof(float);  // ~208 KB (<320 KB/WGP)

__global__ __launch_bounds__(256) void soft_cluster_kernel(
    const float* __restrict__ x,        // [B][R][E]
    const float* __restrict__ centers,  // [K][E]
    float* __restrict__ partial)        // [B][CHUNKS][K][E]
{
  extern __shared__ float lds[];
  float* sC  = lds;
  float* sX0 = lds + K * E;
  float* sX1 = sX0 + TILE_ROWS * E;
  float* sP  = sX1 + TILE_ROWS * E;

  const int b     = blockIdx.x / CHUNKS;
  const int chunk = blockIdx.x % CHUNKS;
  const int tid   = threadIdx.x;
  const int wid   = tid >> 5;   // 8 waves (wave32)
  const int lane  = tid & 31;
  const int ln16  = lane & 15;
  const int hi    = lane >> 4;

  const size_t x_base = ((size_t)b * R + (size_t)chunk * ROWS_PER_CHUNK) * E;

  // ---- Async-stage centers (64KB) and the first x tile (64KB) into LDS ----
  #pragma unroll
  for (int i = 0; i < (K * E / 4) / 256; ++i)
    async_b128(centers + (size_t)(tid + i * 256) * 4, sC + (tid + i * 256) * 4);
  #pragma unroll
  for (int i = 0; i < (TILE_ROWS * E / 4) / 256; ++i)
    async_b128(x + x_base + (size_t)(tid + i * 256) * 4, sX0 + (tid + i * 256) * 4);

  // ---- Persistent z accumulators: wave owns mtile = wid>>1, 8 ntiles ----
  const int mtile  = wid >> 1;
  const int parity = wid & 1;
  v8f acc[8];
  #pragma unroll
  for (int j = 0; j < 8; ++j) acc[j] = (v8f){0.f,0.f,0.f,0.f,0.f,0.f,0.f,0.f};

  wait_async0();
  __syncthreads();

  for (int it = 0; it < TILE_ITERS; ++it) {
    float* sXc = (it & 1) ? sX1 : sX0;   // compute buffer (ready)
    float* sXn = (it & 1) ? sX0 : sX1;   // prefetch buffer

    // ---- Kick off async prefetch of next tile; overlaps all compute below ----
    if (it + 1 < TILE_ITERS) {
      const float* src = x + x_base + (size_t)(it + 1) * TILE_ROWS * E;
      #pragma unroll
      for (int i = 0; i < (TILE_ROWS * E / 4) / 256; ++i)
        async_b128(src + (size_t)(tid + i * 256) * 4, sXn + (tid + i * 256) * 4);
    }

    // ---- GEMM1: scores(64x64) = xtile(64x256) * centers^T(256x64) ----
    {
      const int rstrip = wid & 3;
      const int kt0    = (wid >> 2) * 2;
      const int kt1    = kt0 + 1;
      v8f s0 = (v8f){0.f,0.f,0.f,0.f,0.f,0.f,0.f,0.f};
      v8f s1 = s0;
      const float* arow = &sXc[(rstrip * 16 + ln16) * E + 2 * hi];
      const float* b0r  = &sC[(kt0 * 16 + ln16) * E + 2 * hi];
      const float* b1r  = &sC[(kt1 * 16 + ln16) * E + 2 * hi];
      #pragma unroll 8
      for (int e = 0; e < E; e += 4) {
        v2f a   = *(const v2f*)(arow + e);
        v2f bb0 = *(const v2f*)(b0r + e);
        v2f bb1 = *(const v2f*)(b1r + e);
        s0 = wmma_f32(a, bb0, s0);
        s1 = wmma_f32(a, bb1, s1);
      }
      // D layout: vgpr v -> M = v + 8*hi ; N = ln16
      #pragma unroll
      for (int v = 0; v < 8; ++v) {
        const int r = rstrip * 16 + v + 8 * hi;
        sP[r * PSTR + kt0 * 16 + ln16] = s0[v];
        sP[r * PSTR + kt1 * 16 + ln16] = s1[v];
      }
    }
    __syncthreads();

    // ---- Softmax over K=64 per row (one thread per row) ----
    if (tid < TILE_ROWS) {
      float* row = &sP[tid * PSTR];
      float mx = row[0];
      #pragma unroll 8
      for (int k = 1; k < K; ++k) mx = fmaxf(mx, row[k]);
      float sum = 0.f;
      #pragma unroll 8
      for (int k = 0; k < K; ++k) { float ev = __expf(row[k] - mx); row[k] = ev; sum += ev; }
      const float inv = 1.0f / sum;
      #pragma unroll 8
      for (int k = 0; k < K; ++k) row[k] *= inv;
    }
    __syncthreads();

    // ---- GEMM2: z(64x256) += p^T(64xrows) * xtile(64x256) ----
    #pragma unroll 4
    for (int rb = 0; rb < TILE_ROWS / 4; ++rb) {
      const int r0 = rb * 4 + 2 * hi;
      v2f a;
      a.x = sP[(r0    ) * PSTR + mtile * 16 + ln16];
      a.y = sP[(r0 + 1) * PSTR + mtile * 16 + ln16];
      #pragma unroll
      for (int j = 0; j < 8; ++j) {
        const int nt = j * 2 + parity;
        v2f bb;
        bb.x = sXc[(r0    ) * E + nt * 16 + ln16];
        bb.y = sXc[(r0 + 1) * E + nt * 16 + ln16];
        acc[j] = wmma_f32(a, bb, acc[j]);
      }
    }

    // ---- Close the tile: next buffer's async copies done, then rendezvous ----
    wait_async0();
    __syncthreads();
  }

  // ---- Write this chunk's 64x256 partial (deterministic split-K) ----
  {
    float* pout = partial + ((size_t)b * CHUNKS + chunk) * K * E;
    #pragma unroll
    for (int j = 0; j < 8; ++j) {
      const int nt = j * 2 + parity;
      #pragma unroll
      for (int v = 0; v < 8; ++v) {
        const int kk = mtile * 16 + v + 8 * hi;
        pout[kk * E + nt * 16 + ln16] = acc[j][v];
      }
    }
  }
}

// out[b][k][e] = sum_c partial[b][c][k][e]  (fixed order -> bitwise deterministic)
__global__ __launch_bounds__(256) void reduce_partials_kernel(
    const float* __restrict__ partial, float* __restrict__ out)
{
  const int idx = blockIdx.x * 256 + threadIdx.x;
  if (idx >= Bb * K * E) return;
  const int b   = idx / (K * E);
  const int rem = idx % (K * E);
  const float* p = partial + (size_t)b * CHUNKS * K * E + rem;
  float s = 0.f;
  #pragma unroll
  for (int c = 0; c < CHUNKS; ++c) s += p[(size_t)c * K * E];
  out[idx] = s;
}

extern "C" void kernel_launch(void* const* d_in, const int* in_sizes, int n_in,
                              void* d_out, int out_size, void* d_ws, size_t ws_size,
                              hipStream_t stream) {
  const float* x       = (const float*)d_in[0];   // (B, R, E) f32
  const float* centers = (const float*)d_in[1];   // (K, E)   f32
  float* out     = (float*)d_out;                 // (B, K, E) f32
  float* partial = (float*)d_ws;                  // B*CHUNKS*K*E f32 = 32 MB

  soft_cluster_kernel<<<dim3(Bb * CHUNKS), dim3(256), LDS_BYTES, stream>>>(
      x, centers, partial);

  const int total = Bb * K * E;
  reduce_partials_kernel<<<dim3((total + 255) / 256), dim3(256), 0, stream>>>(
      partial, out);
}